// risk_interaction_89404039233801
// MI455X (gfx1250) — compile-verified
//
#include <hip/hip_runtime.h>
#include <hip/hip_bf16.h>
#include <math.h>

typedef __attribute__((ext_vector_type(2))) float v2f;
typedef __attribute__((ext_vector_type(8))) float v8f;

#define PI_F   3.14159265358979323846f
#define PIO2_F 1.57079632679489661923f

// Workspace layout (per (t,n), SoAoS float4):
//   F0 = {pos.x, pos.y, v, ang}
//   F1 = {cos(ang), sin(ang), node, li}   li = (obs_traj_type==4) ? 0.65 : 1.0

// ---------------------------------------------------------------------------
// Pass 1: per-(t, n) features -> two float4 arrays.
// ---------------------------------------------------------------------------
__global__ void risk_pass1(const float* __restrict__ a,
                           const float* __restrict__ Wn1,
                           const float* __restrict__ Wn2,
                           const float* __restrict__ bn,
                           const float* __restrict__ W5,
                           const float* __restrict__ b5,
                           const int*   __restrict__ otype,
                           float4* __restrict__ F0,
                           float4* __restrict__ F1,
                           int N, int T)
{
    int gid = blockIdx.x * blockDim.x + threadIdx.x;
    int Tm1 = T - 1;
    if (gid >= Tm1 * N) return;
    int n = gid % N;
    int t = gid / N;

    // a layout: (N, 2, T) row-major
    const float* an = a + (size_t)n * 2 * T;
    float pxv = an[t + 1];
    float pyv = an[T + t + 1];
    float sx  = pxv - an[t];
    float sy  = pyv - an[T + t];
    float h    = sqrtf(fmaxf(sx * sx + sy * sy, 1e-18f));
    float vval = h * 2.0f;                 // / DT(=0.5)
    float aval = atan2f(sy, sx);
    float invh = __builtin_amdgcn_rcpf(h);
    float cav  = sx * invh;                // cos(ang)
    float sav  = sy * invh;                // sin(ang)

    // last agent (index N-1) features for neighbor gate
    const float* al = a + (size_t)(N - 1) * 2 * T;
    float pxL = al[t + 1];
    float pyL = al[T + t + 1];
    float sxL = pxL - al[t];
    float syL = pyL - al[T + t];
    float vL  = sqrtf(fmaxf(sxL * sxL + syL * syL, 1e-18f)) * 2.0f;
    float aL  = atan2f(syL, sxL);

    float ddx = pxv - pxL, ddy = pyv - pyL;
    float dis_nbr = sqrtf(fmaxf(ddx * ddx + ddy * ddy, 1e-18f));
    bool nearL = (dis_nbr <= 12.0f);

    float f[4]  = {pxv, pyv, vval, aval};
    float fb[4] = {pxL, pyL, vL,  aL};
    float node = b5[0];
#pragma unroll
    for (int c = 0; c < 4; ++c) {
        float hh = nearL ? (f[c] * Wn2[2 * c] + fb[c] * Wn2[2 * c + 1] + bn[c])
                         : (f[c] * Wn1[c] + bn[c]);
        node += hh * W5[c];
    }
    float li = (otype[n] == 4) ? 0.65f : 1.0f;

    F0[gid] = make_float4(pxv, pyv, vval, aval);
    F1[gid] = make_float4(cav, sav, node, li);
}

// ---------------------------------------------------------------------------
// Pass 2: one wave32 per 16x16 (i,j) tile, no transcendentals.
//   bb tile via V_WMMA_F32_16X16X4_F32 (rank-2: Wr0*node_i + Wr1*node_j + br)
//   Epilogue in the WMMA C/D register layout:
//     lane l -> column j = jBase + (l & 15);  VGPR r -> row i = iBase + r + 8*(l>>4)
//   Cone test: half-plane sign + wrap correction (no atan2).
//   vv/dis collapses: |v_i*(dx*cai+dy*sai) - v_j*(dx*caj+dy*saj)| / max(sq,1e-18)
//   1/x via hardware V_RCP_F32 (~1 ulp).
//   Row addressing: one shared 32-bit BYTE offset VGPR for F0 and F1
//   (GVS: SGPR base + VGPR byteoff + IOFFSET; the unrolled +16*r folds into
//   the instruction immediate). Diagonal handled by select, not branch, so
//   the loop body is branch-free and clauseable.
// Grid is sized exactly; EXEC all-1s at the WMMA.
// ---------------------------------------------------------------------------
__global__ void risk_pass2(const float4* __restrict__ F0,
                           const float4* __restrict__ F1,
                           const float* __restrict__ Wr, const float* __restrict__ brp,
                           float* __restrict__ out,
                           int N)
{
    const unsigned uN = (unsigned)N;
    const int tilesPerDim = N >> 4;
    const int tilesPerT   = tilesPerDim * tilesPerDim;
    const int wavesPerBlk = blockDim.x >> 5;
    const int wave = threadIdx.x >> 5;
    const int lane = threadIdx.x & 31;

    const int tile = blockIdx.x * wavesPerBlk + wave;   // grid sized exactly
    const int t    = tile / tilesPerT;
    const int rem  = tile - t * tilesPerT;
    const int iT   = rem / tilesPerDim;
    const int jT   = rem - iT * tilesPerDim;

    const int jn   = lane & 15;
    const int half = lane >> 4;
    const int jBase = jT << 4;
    const int iBase = iT << 4;
    const int j = jBase + jn;

    const unsigned tN = (unsigned)t * uN;

    const float Wr0 = Wr[0], Wr1 = Wr[1], br = brp[0];

    // column-j features (one per lane; lanes 16..31 mirror lanes 0..15)
    const unsigned jIdx = tN + (unsigned)j;
    const float4 f0j = F0[jIdx];   // px, py, v, (ang unused)
    const float4 f1j = F1[jIdx];   // ca, sa, node, (li unused)
    const float pxj = f0j.x, pyj = f0j.y, vj = f0j.z;
    const float caj = f1j.x, saj = f1j.y;
    const float nodej = f1j.z;

    // A (16x4 f32): lanes 0-15 hold {K0,K1} for M=lane; lanes 16-31 hold {K2,K3}.
    //   A[m][0] = Wr0*node_i[m], A[m][1] = 1, A[m][2..3] = 0.
    const float nodeiA = F1[tN + (unsigned)(iBase + jn)].z;
    v2f A;
    A.x = (half == 0) ? Wr0 * nodeiA : 0.0f;
    A.y = (half == 0) ? 1.0f         : 0.0f;
    // B (4x16 f32): lanes 0-15 hold {K0,K1} for N=lane; lanes 16-31 hold {K2,K3}.
    //   B[0][n] = 1, B[1][n] = Wr1*node_j[n], B[2..3][n] = 0.
    v2f B;
    B.x = (half == 0) ? 1.0f         : 0.0f;
    B.y = (half == 0) ? Wr1 * nodej  : 0.0f;

    v8f C;
#pragma unroll
    for (int r = 0; r < 8; ++r) C[r] = br;

    // bb[m][n] = Wr0*node_i[m] + Wr1*node_j[n] + br   (D = A x B + C)
    v8f bb = __builtin_amdgcn_wmma_f32_16x16x4_f32(
        /*neg_a=*/false, A, /*neg_b=*/false, B,
        /*c_mod=*/(short)0, C, /*reuse_a=*/false, /*reuse_b=*/false);

    // Shared byte offset for the 8 row records of F0/F1; +16*r folds into the
    // load immediates. Output index advances by one 32-bit add per row.
    const unsigned rowBase  = tN + (unsigned)(iBase + (half << 3));
    const unsigned rowBytes = rowBase * 16u;
    const char* F0b = (const char*)F0;
    const char* F1b = (const char*)F1;
    unsigned oIdx = rowBase * uN + (unsigned)j;
    const int iFirst = iBase + (half << 3);

#pragma unroll
    for (int r = 0; r < 8; ++r) {
        const int i = iFirst + r;

        const float4 f0i = *(const float4*)(F0b + rowBytes + 16u * (unsigned)r);
        const float4 f1i = *(const float4*)(F1b + rowBytes + 16u * (unsigned)r);
        const float pxi = f0i.x, pyi = f0i.y, vi = f0i.z, angi = f0i.w;
        const float cai = f1i.x, sai = f1i.y, li = f1i.w;

        // diff[t,i,j] = pos[j] - pos[i]
        const float dx = pxj - pxi;
        const float dy = pyj - pyi;
        const float sq = fmaf(dx, dx, dy * dy);

        // unnormalized cos(ang - angle3) * r
        const float ciu = fmaf(dx, cai, dy * sai);
        const float cju = fmaf(dx, caj, dy * saj);

        // cone: half-plane + wrap correction (no atan2); diagonal folded in
        const bool okPos = (angi >  PIO2_F) ? (dy >= 0.0f) : true;
        const bool okNeg = (angi < -PIO2_F) ? (dy <  0.0f) : true;
        const bool gate  = (ciu > 0.0f) && okPos && okNeg && (i != j);
        const float lij  = gate ? li : 0.0f;

        // vv/dis = |vi*ciu - vj*cju| / (r*dis) ; r*dis == max(sq, 1e-18)
        const float num    = fabsf(vi * ciu - vj * cju);
        const float factor = __builtin_amdgcn_rcpf(fmaxf(sq, 1e-18f));

        out[oIdx] = num * factor * bb[r] * lij;
        oIdx += uN;
    }
}

// ---------------------------------------------------------------------------
// Host launcher
// ---------------------------------------------------------------------------
extern "C" void kernel_launch(void* const* d_in, const int* in_sizes, int n_in,
                              void* d_out, int out_size, void* d_ws, size_t ws_size,
                              hipStream_t stream) {
    // setup_inputs order:
    // 0:a 1:sa_out 2:se_out 3:Wn1 4:Wn2 5:bn 6:W5 7:b5 8:Wr 9:br
    // 10:pedestrian_index 11:obs_traj_type 12:start 13:end
    const float* a   = (const float*)d_in[0];
    const float* Wn1 = (const float*)d_in[3];
    const float* Wn2 = (const float*)d_in[4];
    const float* bn  = (const float*)d_in[5];
    const float* W5  = (const float*)d_in[6];
    const float* b5  = (const float*)d_in[7];
    const float* Wr  = (const float*)d_in[8];
    const float* br  = (const float*)d_in[9];
    const int* otype = (const int*)d_in[11];

    const int N   = in_sizes[11];          // 512
    const int T   = in_sizes[0] / (2 * N); // 20
    const int Tm1 = T - 1;
    const int S   = Tm1 * N;               // 9728

    float4* F0 = (float4*)d_ws;
    float4* F1 = F0 + (size_t)S;

    const int thr1 = 256;
    const int blk1 = (S + thr1 - 1) / thr1;
    risk_pass1<<<blk1, thr1, 0, stream>>>(a, Wn1, Wn2, bn, W5, b5, otype,
                                          F0, F1, N, T);

    const int tilesPerDim = N / 16;
    const int tiles = Tm1 * tilesPerDim * tilesPerDim;  // 19456
    const int wavesPerBlk = (tiles % 8 == 0) ? 8 : 1;   // exact grid -> EXEC all-1s
    const int blk2 = wavesPerBlk * 32;
    const int grd2 = tiles / wavesPerBlk;
    risk_pass2<<<grd2, blk2, 0, stream>>>(F0, F1, Wr, br, (float*)d_out, N);
}